// InverseTransformationBilinear_1589137899538
// MI455X (gfx1250) — compile-verified
//
#include <hip/hip_runtime.h>
#include <cstdint>

// Problem constants (from reference): img (3,512,512) f32, theta (64,) f32,
// out (64,3,512,512) f32.
#define CCH 3
#define HH  512
#define WW  512
#define NN  64
#define TS  32      // output tile side
#define LDSP 48     // LDS row pitch (floats); box side <= 48 after widening
#define LDSR 48

typedef float v4f __attribute__((ext_vector_type(4)));

typedef __attribute__((address_space(1))) int* as1i;   // global int*
typedef __attribute__((address_space(3))) int* as3i;   // LDS int*

__device__ __forceinline__ void async_copy_f32(const float* gsrc, float* ldst) {
#if __has_builtin(__builtin_amdgcn_global_load_async_to_lds_b32)
  // global -> LDS DMA-style copy, tracked by ASYNCcnt (CDNA5 path)
  __builtin_amdgcn_global_load_async_to_lds_b32(
      (as1i)(uintptr_t)gsrc,
      (as3i)(uint32_t)(uintptr_t)ldst,   // generic LDS ptr low 32 bits == AS3 offset
      0, 0);
#else
  *ldst = *gsrc;                         // sync fallback (still correct)
#endif
}

__device__ __forceinline__ void wait_async0() {
#if __has_builtin(__builtin_amdgcn_s_wait_asynccnt)
  __builtin_amdgcn_s_wait_asynccnt(0);
#else
  asm volatile("s_wait_asynccnt 0" ::: "memory");
#endif
}

__global__ __launch_bounds__(256)
void rot_bilinear_kernel(const float* __restrict__ img,
                         const float* __restrict__ theta,
                         float* __restrict__ out) {
  __shared__ float smem[CCH][LDSR * LDSP];   // 27,648 B

  const int n      = blockIdx.y;
  const int tileid = blockIdx.x;             // 0..255
  const int tx     = (tileid & 15) * TS;
  const int ty     = (tileid >> 4) * TS;
  const int tid    = threadIdx.x;

  float s, c;
  sincosf(theta[n], &s, &c);

  // Normalized coords exactly like the reference: xn = -1 + 2*x/511,
  // ix = (c*xn - s*yn + 1) * 255.5, iy = (s*xn + c*yn + 1) * 255.5
  const float inv = 2.0f / 511.0f;
  const float fxa = (float)tx * inv - 1.0f;
  const float fxb = (float)(tx + TS - 1) * inv - 1.0f;
  const float fya = (float)ty * inv - 1.0f;
  const float fyb = (float)(ty + TS - 1) * inv - 1.0f;

  // Source-space bounding box of this output tile (affine => extremes at corners)
  const float ix0c = (c * fxa - s * fya + 1.0f) * 255.5f;
  const float ix1c = (c * fxb - s * fya + 1.0f) * 255.5f;
  const float ix2c = (c * fxa - s * fyb + 1.0f) * 255.5f;
  const float ix3c = (c * fxb - s * fyb + 1.0f) * 255.5f;
  const float iy0c = (s * fxa + c * fya + 1.0f) * 255.5f;
  const float iy1c = (s * fxb + c * fya + 1.0f) * 255.5f;
  const float iy2c = (s * fxa + c * fyb + 1.0f) * 255.5f;
  const float iy3c = (s * fxb + c * fyb + 1.0f) * 255.5f;

  const float ixmin = fminf(fminf(ix0c, ix1c), fminf(ix2c, ix3c));
  const float ixmax = fmaxf(fmaxf(ix0c, ix1c), fmaxf(ix2c, ix3c));
  const float iymin = fminf(fminf(iy0c, iy1c), fminf(iy2c, iy3c));
  const float iymax = fmaxf(fmaxf(iy0c, iy1c), fmaxf(iy2c, iy3c));

  // widen by 1 each side (+1 neighbor + FP safety); clamp to image
  int bx0 = (int)floorf(ixmin) - 1;
  int bx1 = (int)floorf(ixmax) + 2;
  int by0 = (int)floorf(iymin) - 1;
  int by1 = (int)floorf(iymax) + 2;
  const int bx0c = max(bx0, 0), bx1c = min(bx1, WW - 1);
  const int by0c = max(by0, 0), by1c = min(by1, HH - 1);
  const int boxW = bx1c - bx0c + 1;          // <= 48
  const int boxH = by1c - by0c + 1;          // <= 48

  // ---- Stage bounding box (3 channels) into LDS via async copies ----
  const int per   = boxW * boxH;
  const int total = CCH * per;               // may be <= 0 if tile fully outside
  for (int i = tid; i < total; i += 256) {
    const int ch  = i / per;
    const int rem = i - ch * per;
    const int r   = rem / boxW;
    const int col = rem - r * boxW;
    const float* g = img + ch * (HH * WW) + (by0c + r) * WW + (bx0c + col);
    async_copy_f32(g, &smem[ch][r * LDSP + col]);
  }
  wait_async0();
  __syncthreads();

  // ---- Interpolate: thread -> 4 consecutive x, all 3 channels ----
  const int lrow  = tid >> 3;                // 0..31
  const int lcol4 = (tid & 7) * 4;           // 0,4,...,28
  const int oy    = ty + lrow;
  const float fy  = (float)oy * inv - 1.0f;

  float acc[CCH][4];
#pragma unroll
  for (int j = 0; j < 4; ++j) {
    const int   ox = tx + lcol4 + j;
    const float fx = (float)ox * inv - 1.0f;
    const float ix = (c * fx - s * fy + 1.0f) * 255.5f;
    const float iy = (s * fx + c * fy + 1.0f) * 255.5f;

    const float xf0 = floorf(ix), yf0 = floorf(iy);
    const float wx = ix - xf0,    wy = iy - yf0;
    const float xf1 = xf0 + 1.0f, yf1 = yf0 + 1.0f;

    const bool mx0 = (xf0 >= 0.0f) && (xf0 <= 511.0f);
    const bool mx1 = (xf1 >= 0.0f) && (xf1 <= 511.0f);
    const bool my0 = (yf0 >= 0.0f) && (yf0 <= 511.0f);
    const bool my1 = (yf1 >= 0.0f) && (yf1 <= 511.0f);

    const float w00 = (1.0f - wx) * (1.0f - wy);
    const float w01 = wx * (1.0f - wy);
    const float w10 = (1.0f - wx) * wy;
    const float w11 = wx * wy;

    const int xi0 = min(max((int)xf0, 0), WW - 1) - bx0c;
    const int xi1 = min(max((int)xf1, 0), WW - 1) - bx0c;
    const int yi0 = min(max((int)yf0, 0), HH - 1) - by0c;
    const int yi1 = min(max((int)yf1, 0), HH - 1) - by0c;

    const int i00 = yi0 * LDSP + xi0;
    const int i01 = yi0 * LDSP + xi1;
    const int i10 = yi1 * LDSP + xi0;
    const int i11 = yi1 * LDSP + xi1;

#pragma unroll
    for (int ch = 0; ch < CCH; ++ch) {
      // select-to-zero BEFORE weighting (matches reference mask; avoids NaN
      // from unstaged LDS when the tile maps fully outside the image)
      const float v00 = (mx0 && my0) ? smem[ch][i00] : 0.0f;
      const float v01 = (mx1 && my0) ? smem[ch][i01] : 0.0f;
      const float v10 = (mx0 && my1) ? smem[ch][i10] : 0.0f;
      const float v11 = (mx1 && my1) ? smem[ch][i11] : 0.0f;
      acc[ch][j] = v00 * w00 + v01 * w01 + v10 * w10 + v11 * w11;
    }
  }

  // ---- Coalesced 16B non-temporal streaming stores (out is write-once) ----
#pragma unroll
  for (int ch = 0; ch < CCH; ++ch) {
    v4f v = { acc[ch][0], acc[ch][1], acc[ch][2], acc[ch][3] };
    float* dst = out + (((size_t)n * CCH + ch) * HH + oy) * WW + (tx + lcol4);
    __builtin_nontemporal_store(v, (v4f*)dst);
  }
}

extern "C" void kernel_launch(void* const* d_in, const int* in_sizes, int n_in,
                              void* d_out, int out_size, void* d_ws, size_t ws_size,
                              hipStream_t stream) {
  const float* img   = (const float*)d_in[0];   // (3,512,512) f32
  const float* theta = (const float*)d_in[1];   // (64,) f32
  float* out = (float*)d_out;                   // (64,3,512,512) f32

  dim3 grid((WW / TS) * (HH / TS), NN);         // 256 tiles x 64 angles
  rot_bilinear_kernel<<<grid, 256, 0, stream>>>(img, theta, out);
}